// AttentionTeacherAlignment_30897994727733
// MI455X (gfx1250) — compile-verified
//
#include <hip/hip_runtime.h>

// Problem sizes (fixed by the reference).
#define T_ 128
#define B_ 512
#define S_ 512
#define F_ 8

// WMMA fragment types for V_WMMA_F32_16X16X4_F32:
// A (16x4 f32) = 64 vals / 32 lanes = 2 VGPRs -> v2f ; C/D (16x16 f32) = v8f.
typedef float v2f __attribute__((ext_vector_type(2)));
typedef float v8f __attribute__((ext_vector_type(8)));

#define ROWSTRIDE 516  // 512 + 4 dword pad: (l*516)%64 == 4l -> conflict-free b64 reads

// One workgroup = one (batch b, 16-row t-tile). 256 threads = 8 waves.
// Wave w reduces s in [w*64, w*64+64) with 16 chained WMMAs.
__global__ __launch_bounds__(256) void attn_align_main(
    const float* __restrict__ att,    // [T,B,S]
    const float* __restrict__ gates,  // [F,B,T]
    const int*   __restrict__ mrs,    // [B,S]
    const int*   __restrict__ fmap,   // [V]
    float*       __restrict__ partials)  // [4096]
{
    __shared__ __align__(16) float sA[16 * ROWSTRIDE];  // 33 KB attention tile
    __shared__ float    sP[8 * 256];                    // per-wave 16x16 P tiles
    __shared__ unsigned s_fidx4[S_ / 4];                // fidx packed 4 x u8 per dword
    __shared__ unsigned s_cnt[F_ + 1];                  // histogram counts
    __shared__ float    s_red[16];
    __shared__ float    s_a2t[256];
    __shared__ float    s_a2s[64];

    const int tid = threadIdx.x;
    const int wg  = blockIdx.x;
    const int b   = wg >> 3;          // batch
    const int t0  = (wg & 7) << 4;    // t-tile base

    if (tid <= F_) s_cnt[tid] = 0u;
    __syncthreads();

    // ---- fidx = field_map[mrs[b,s]] (packed u8) + per-batch histogram ----
    unsigned char* s_fidx = (unsigned char*)s_fidx4;
    for (int s = tid; s < S_; s += 256) {
        const int f = fmap[mrs[b * S_ + s]];            // 0..8
        s_fidx[s] = (unsigned char)f;
        atomicAdd(&s_cnt[f], 1u);
    }

    // ---- stage attention rows t0..t0+15 (all 512 s) into LDS; fuse sum(a^2) ----
    float a2 = 0.f;
#pragma unroll
    for (int k = 0; k < 8; ++k) {
        const int j  = tid + (k << 8);                  // 0..2047 float4 tiles
        const int m  = j >> 7;                          // row 0..15
        const int s4 = j & 127;                         // float4 col
        const float4 v = *reinterpret_cast<const float4*>(
            att + ((size_t)(t0 + m) * B_ + b) * S_ + ((size_t)s4 << 2));
        a2 += v.x * v.x + v.y * v.y + v.z * v.z + v.w * v.w;
        *reinterpret_cast<float4*>(&sA[m * ROWSTRIDE + (s4 << 2)]) = v;
    }
    s_a2t[tid] = a2;
    __syncthreads();

    // ---- one-hot GEMM: P[t,f] += sum_s a[t,s] * (fidx[s]==f) via WMMA ----
    const int lane = tid & 31;
    const int wv   = tid >> 5;        // wave id 0..7
    const int n    = lane & 15;       // N index (field)
    const int hi   = lane >> 4;       // half-wave: K offset 0 / 2
    const int rowb = n * ROWSTRIDE;   // A row = lane&15

    v8f c = {0.f, 0.f, 0.f, 0.f, 0.f, 0.f, 0.f, 0.f};
#pragma unroll
    for (int q = 0; q < 16; ++q) {
        const int s0 = (wv << 6) + (q << 2);
        const unsigned w = s_fidx4[s0 >> 2];            // broadcast LDS read
        // B fragment (4x16 one-hot): V0 <-> K=2*hi, V1 <-> K=2*hi+1
        const unsigned f0 = (w >> (16 * hi)) & 0xFFu;
        const unsigned f1 = (w >> (16 * hi + 8)) & 0xFFu;
        v2f bf;
        bf[0] = (f0 == (unsigned)n) ? 1.0f : 0.0f;
        bf[1] = (f1 == (unsigned)n) ? 1.0f : 0.0f;
        // A fragment: row = lane&15, cols s0+2*hi, s0+2*hi+1 (conflict-free b64)
        const float2 av = *reinterpret_cast<const float2*>(&sA[rowb + s0 + 2 * hi]);
        v2f af; af[0] = av.x; af[1] = av.y;
        c = __builtin_amdgcn_wmma_f32_16x16x4_f32(
            /*neg_a=*/false, af, /*neg_b=*/false, bf,
            /*c_mod=*/(short)0, c, /*reuse_a=*/false, /*reuse_b=*/false);
    }

    // ---- spill per-wave C (deterministic fixed-order combine later) ----
    // C/D layout: VGPR j, lanes0-15 -> M=j, N=lane ; lanes16-31 -> M=8+j
#pragma unroll
    for (int j = 0; j < 8; ++j)
        sP[(wv << 8) + ((j + (hi << 3)) << 4) + n] = c[j];
    __syncthreads();

    // ---- per-(t,b) loss terms (threads 0..15), a^2 stage-1 (threads 0..63) ----
    if (tid < 16) {
        const int t = t0 + tid;
        float nrm = 0.f, t1 = 0.f, t2 = 0.f;
#pragma unroll
        for (int f = 1; f <= F_; ++f) {
            float Pf = 0.f;
#pragma unroll
            for (int w2 = 0; w2 < 8; ++w2)             // fixed-order wave combine
                Pf += sP[(w2 << 8) + (tid << 4) + f];
            const float g  = gates[((size_t)(f - 1) * B_ + b) * T_ + t];
            const float cf = (float)s_cnt[f];
            nrm += cf * g;
            t1  += cf * g * g;
            t2  += g * Pf;
        }
        const float nn = (nrm == 0.f) ? 1.f : nrm;     // matches jnp.where(norm==0,1,norm)
        s_red[tid] = t1 / (nn * nn) - 2.f * t2 / nn;
    }
    if (tid < 64)
        s_a2s[tid] = s_a2t[tid] + s_a2t[tid + 64] + s_a2t[tid + 128] + s_a2t[tid + 192];
    __syncthreads();

    if (tid == 0) {
        float tot = 0.f;
        for (int i = 0; i < 16; ++i) tot += s_red[i];
        for (int i = 0; i < 64; ++i) tot += s_a2s[i];
        partials[wg] = tot;                             // plain store, no atomics
    }
}

// Deterministic fixed-order final reduction of 4096 partials (double precision).
__global__ __launch_bounds__(256) void attn_align_finalize(
    const float* __restrict__ partials, float* __restrict__ out)
{
    __shared__ double sd[256];
    double acc = 0.0;
#pragma unroll
    for (int k = 0; k < 16; ++k)
        acc += (double)partials[threadIdx.x + (k << 8)];
    sd[threadIdx.x] = acc;
    __syncthreads();
    if (threadIdx.x == 0) {
        double tot = 0.0;
        for (int i = 0; i < 256; ++i) tot += sd[i];
        out[0] = (float)(tot / (double)((long long)T_ * B_ * S_));
    }
}

extern "C" void kernel_launch(void* const* d_in, const int* in_sizes, int n_in,
                              void* d_out, int out_size, void* d_ws, size_t ws_size,
                              hipStream_t stream) {
    (void)in_sizes; (void)n_in; (void)out_size; (void)ws_size;
    const float* att   = (const float*)d_in[0];  // [T,B,S] f32
    const float* gates = (const float*)d_in[1];  // [F,B,T] f32
    const int*   mrs   = (const int*)d_in[2];    // [B,S] i32
    const int*   fmap  = (const int*)d_in[3];    // [V] i32
    float* partials = (float*)d_ws;              // 4096 floats (16 KB)

    attn_align_main<<<dim3(4096), dim3(256), 0, stream>>>(att, gates, mrs, fmap, partials);
    attn_align_finalize<<<dim3(1), dim3(256), 0, stream>>>(partials, (float*)d_out);
}